// LNCLIPDF_29051158790349
// MI455X (gfx1250) — compile-verified
//
#include <hip/hip_runtime.h>
#include <math.h>

#define B_N      2048
#define D_DIM    768
#define NUM_AUG  6144
#define N_ALL    8192
#define EPS_F    1e-6f

typedef __attribute__((ext_vector_type(2))) float v2f;
typedef __attribute__((ext_vector_type(8))) float v8f;

// ---------------------------------------------------------------- init
__global__ void k_init_acc(float* acc) {
    if (threadIdx.x < 8) acc[threadIdx.x] = 0.0f;
}

// ------------------------------------------------- partner stats (per i)
__global__ void k_partner_stats(const int* __restrict__ labels,
                                int* __restrict__ counts,
                                int* __restrict__ rank_,
                                int* __restrict__ start_) {
    int i = blockIdx.x * blockDim.x + threadIdx.x;
    if (i >= B_N) return;
    int li = labels[i];
    int c = 0, r = 0, s = 0;
    for (int j = 0; j < B_N; ++j) {
        int lj = labels[j];
        c += (lj == li);
        r += (lj == li && j < i);
        s += (lj < li);
    }
    counts[i] = c; rank_[i] = r; start_[i] = s;
}

// -------------------------------------------------------------- scatter
__global__ void k_scatter(const int* __restrict__ rank_,
                          const int* __restrict__ start_,
                          int* __restrict__ s2o) {
    int i = blockIdx.x * blockDim.x + threadIdx.x;
    if (i >= B_N) return;
    s2o[start_[i] + rank_[i]] = i;
}

// ------------------------------------------- j-index selection + y_all
__global__ void k_jsel(const int* __restrict__ labels,
                       const int* __restrict__ i_idx,
                       const float* __restrict__ u_j,
                       const int* __restrict__ counts,
                       const int* __restrict__ rank_,
                       const int* __restrict__ start_,
                       const int* __restrict__ s2o,
                       int* __restrict__ jidx,
                       int* __restrict__ yall) {
    int t = blockIdx.x * blockDim.x + threadIdx.x;
    if (t < B_N) yall[t] = labels[t];
    if (t < NUM_AUG) {
        int i   = i_idx[t];
        int cnt = counts[i];
        int r   = (int)floorf(u_j[t] * (float)(cnt - 1));
        int hi  = max(cnt - 2, 0);
        r = min(max(r, 0), hi);
        int jr = (rank_[i] + 1 + r) % max(cnt, 1);
        int j  = s2o[start_[i] + jr];
        jidx[t] = j;
        yall[B_N + t] = labels[i];
    }
}

// ------------------------------------------------------ row normalize
__global__ __launch_bounds__(256) void k_normalize(const float* __restrict__ feats,
                                                   float* __restrict__ zall,
                                                   float* __restrict__ sqn) {
    __shared__ float red[256];
    int row = blockIdx.x;
    const float* f = feats + (size_t)row * D_DIM;
    float s = 0.0f;
    for (int k = threadIdx.x; k < D_DIM; k += 256) { float v = f[k]; s += v * v; }
    red[threadIdx.x] = s; __syncthreads();
    for (int off = 128; off; off >>= 1) {
        if (threadIdx.x < off) red[threadIdx.x] += red[threadIdx.x + off];
        __syncthreads();
    }
    float ss  = red[0];
    float inv = 1.0f / fmaxf(sqrtf(ss), 1e-12f);
    for (int k = threadIdx.x; k < D_DIM; k += 256)
        zall[(size_t)row * D_DIM + k] = f[k] * inv;
    if (threadIdx.x == 0) sqn[row] = ss * inv * inv;
}

// ---------------------------------------------------------- slerp rows
__global__ __launch_bounds__(256) void k_slerp(const float* __restrict__ t_interp,
                                               const int* __restrict__ i_idx,
                                               const int* __restrict__ jidx,
                                               float* __restrict__ zall,
                                               float* __restrict__ sqn) {
    __shared__ float red[256];
    __shared__ float sh_dot, sh_s2;
    int a = blockIdx.x;
    int i = i_idx[a], j = jidx[a];
    const float* zi = zall + (size_t)i * D_DIM;
    const float* zj = zall + (size_t)j * D_DIM;
    float vi[3], vj[3];
    float d = 0.0f;
    #pragma unroll
    for (int e = 0; e < 3; ++e) {
        int k = threadIdx.x + e * 256;
        vi[e] = zi[k]; vj[e] = zj[k];
        d += vi[e] * vj[e];
    }
    red[threadIdx.x] = d; __syncthreads();
    for (int off = 128; off; off >>= 1) {
        if (threadIdx.x < off) red[threadIdx.x] += red[threadIdx.x + off];
        __syncthreads();
    }
    if (threadIdx.x == 0) sh_dot = red[0];
    __syncthreads();
    float cost  = fminf(fmaxf(sh_dot, -1.0f + EPS_F), 1.0f - EPS_F);
    float theta = acosf(cost);
    float sint  = sinf(theta);
    float t     = t_interp[a];
    float ci, cj;
    if (sint < EPS_F) { ci = 1.0f - t; cj = t; }
    else { ci = sinf((1.0f - t) * theta) / sint; cj = sinf(t * theta) / sint; }
    float v[3]; float s2 = 0.0f;
    #pragma unroll
    for (int e = 0; e < 3; ++e) { v[e] = ci * vi[e] + cj * vj[e]; s2 += v[e] * v[e]; }
    red[threadIdx.x] = s2; __syncthreads();
    for (int off = 128; off; off >>= 1) {
        if (threadIdx.x < off) red[threadIdx.x] += red[threadIdx.x + off];
        __syncthreads();
    }
    if (threadIdx.x == 0) sh_s2 = red[0];
    __syncthreads();
    float tot = sh_s2;
    float inv = 1.0f / fmaxf(sqrtf(tot), 1e-12f);
    #pragma unroll
    for (int e = 0; e < 3; ++e)
        zall[(size_t)(B_N + a) * D_DIM + threadIdx.x + e * 256] = v[e] * inv;
    if (threadIdx.x == 0) sqn[B_N + a] = tot * inv * inv;
}

// ----------------------------------------------------------- CE (wave/row)
__global__ __launch_bounds__(256) void k_ce(const float* __restrict__ zall,
                                            const int* __restrict__ yall,
                                            const float* __restrict__ W,
                                            const float* __restrict__ b,
                                            float* __restrict__ acc) {
    int gw   = (blockIdx.x * blockDim.x + threadIdx.x) >> 5;
    int lane = threadIdx.x & 31;
    if (gw >= N_ALL) return;
    const float* z = zall + (size_t)gw * D_DIM;
    float l0 = 0.0f, l1 = 0.0f;
    for (int k = lane; k < D_DIM; k += 32) {
        float zv = z[k];
        l0 += zv * W[k * 2 + 0];
        l1 += zv * W[k * 2 + 1];
    }
    for (int off = 16; off; off >>= 1) {
        l0 += __shfl_down(l0, off);
        l1 += __shfl_down(l1, off);
    }
    if (lane == 0) {
        l0 += b[0]; l1 += b[1];
        float mx  = fmaxf(l0, l1);
        float lse = mx + logf(expf(l0 - mx) + expf(l1 - mx));
        int   y   = yall[gw];
        float ly  = (y == 0) ? l0 : l1;
        atomicAdd(&acc[3], lse - ly);
    }
}

// --------------------------- fused Gram (WMMA f32) + triangle reductions
__global__ __launch_bounds__(256) void k_gram(const float* __restrict__ zall,
                                              const float* __restrict__ sqn,
                                              const int* __restrict__ yall,
                                              float* __restrict__ acc) {
    const int ib = blockIdx.y * 32;   // I rows: ib .. ib+31
    const int jb = blockIdx.x * 64;   // J rows: jb .. jb+63
    if (jb + 63 <= ib) return;        // entirely below strict-upper triangle

    __shared__ __attribute__((aligned(16))) float sI[32 * 66];
    __shared__ __attribute__((aligned(16))) float sJ[64 * 66];
    __shared__ float bacc[3];

    const int tid  = threadIdx.x;
    if (tid < 3) bacc[tid] = 0.0f;
    const int lane = tid & 31;
    const int w    = tid >> 5;
    const int wi   = w & 1;                 // I sub-tile (0..1)
    const int wj   = w >> 1;                // J sub-tile (0..3)
    const int m    = lane & 15;
    const int half = lane >> 4;
    const int koff = half << 1;             // 0 or 2

    v8f accv = {};
    for (int k0 = 0; k0 < D_DIM; k0 += 64) {
        __syncthreads();
        for (int idx = tid; idx < 32 * 64; idx += 256) {
            int r = idx >> 6, c = idx & 63;
            sI[r * 66 + c] = zall[(size_t)(ib + r) * D_DIM + k0 + c];
        }
        for (int idx = tid; idx < 64 * 64; idx += 256) {
            int r = idx >> 6, c = idx & 63;
            sJ[r * 66 + c] = zall[(size_t)(jb + r) * D_DIM + k0 + c];
        }
        __syncthreads();
        const float* pA = &sI[(wi * 16 + m) * 66 + koff];
        const float* pB = &sJ[(wj * 16 + m) * 66 + koff];
        #pragma unroll
        for (int kk = 0; kk < 64; kk += 4) {
            v2f av = *(const v2f*)(pA + kk);
            v2f bv = *(const v2f*)(pB + kk);
            accv = __builtin_amdgcn_wmma_f32_16x16x4_f32(
                false, av, false, bv, (short)0, accv, false, false);
        }
    }

    // Epilogue: D tile -> masked reductions
    float uni = 0.0f, s0 = 0.0f, s1 = 0.0f;
    #pragma unroll
    for (int r = 0; r < 8; ++r) {
        int gi = ib + wi * 16 + r + half * 8;
        int gj = jb + wj * 16 + m;
        if (gj > gi) {
            float g  = accv[r];
            float d2 = fmaxf(sqn[gi] + sqn[gj] - 2.0f * g, 0.0f);
            uni += expf(-2.0f * d2);
            int yi = yall[gi], yj = yall[gj];
            if (yi == yj) { if (yi == 0) s0 += d2; else s1 += d2; }
        }
    }
    for (int off = 16; off; off >>= 1) {
        uni += __shfl_down(uni, off);
        s0  += __shfl_down(s0,  off);
        s1  += __shfl_down(s1,  off);
    }
    if (lane == 0) {
        atomicAdd(&bacc[0], uni);
        atomicAdd(&bacc[1], s0);
        atomicAdd(&bacc[2], s1);
    }
    __syncthreads();
    if (tid == 0) {
        atomicAdd(&acc[0], bacc[0]);
        atomicAdd(&acc[1], bacc[1]);
        atomicAdd(&acc[2], bacc[2]);
    }
}

// -------------------------------------------------------------- combine
__global__ __launch_bounds__(256) void k_combine(const float* __restrict__ acc,
                                                 const int* __restrict__ yall,
                                                 float* __restrict__ out) {
    __shared__ int c0s[256], c1s[256];
    int c0 = 0, c1 = 0;
    for (int t = threadIdx.x; t < N_ALL; t += 256) {
        int y = yall[t];
        c0 += (y == 0); c1 += (y == 1);
    }
    c0s[threadIdx.x] = c0; c1s[threadIdx.x] = c1; __syncthreads();
    for (int off = 128; off; off >>= 1) {
        if (threadIdx.x < off) {
            c0s[threadIdx.x] += c0s[threadIdx.x + off];
            c1s[threadIdx.x] += c1s[threadIdx.x + off];
        }
        __syncthreads();
    }
    if (threadIdx.x == 0) {
        float n0 = (float)c0s[0], n1 = (float)c1s[0];
        float Nf = (float)N_ALL;
        float npairs = Nf * (Nf - 1.0f) * 0.5f;
        float uniform = logf(acc[0] / npairs);
        float p0 = n0 * (n0 - 1.0f) * 0.5f;
        float p1 = n1 * (n1 - 1.0f) * 0.5f;
        float at0 = acc[1] / fmaxf(p0, 1.0f);
        float at1 = acc[2] / fmaxf(p1, 1.0f);
        float v0 = (c0s[0] >= 2) ? 1.0f : 0.0f;
        float v1 = (c1s[0] >= 2) ? 1.0f : 0.0f;
        float align_ = (at0 * v0 + at1 * v1) / fmaxf(v0 + v1, 1.0f);
        float ce = acc[3] / Nf;
        out[0] = ce + 0.1f * align_ + 0.5f * uniform;
    }
}

// ---------------------------------------------------------------- launch
extern "C" void kernel_launch(void* const* d_in, const int* in_sizes, int n_in,
                              void* d_out, int out_size, void* d_ws, size_t ws_size,
                              hipStream_t stream) {
    const float* feats    = (const float*)d_in[0];
    const float* W        = (const float*)d_in[1];
    const float* b        = (const float*)d_in[2];
    const float* u_j      = (const float*)d_in[3];
    const float* t_interp = (const float*)d_in[4];
    const int*   labels   = (const int*)d_in[5];
    const int*   i_idx    = (const int*)d_in[6];

    float* ws     = (float*)d_ws;
    float* zall   = ws;                               // 8192*768 floats
    float* sqn    = zall + (size_t)N_ALL * D_DIM;     // 8192
    float* acc    = sqn + N_ALL;                      // 8
    int*   counts = (int*)(acc + 8);                  // 2048
    int*   rank_  = counts + B_N;                     // 2048
    int*   start_ = rank_ + B_N;                      // 2048
    int*   s2o    = start_ + B_N;                     // 2048
    int*   jidx   = s2o + B_N;                        // 6144
    int*   yall   = jidx + NUM_AUG;                   // 8192

    float* out = (float*)d_out;

    k_init_acc<<<1, 32, 0, stream>>>(acc);
    k_partner_stats<<<(B_N + 255) / 256, 256, 0, stream>>>(labels, counts, rank_, start_);
    k_scatter<<<(B_N + 255) / 256, 256, 0, stream>>>(rank_, start_, s2o);
    k_jsel<<<(N_ALL + 255) / 256, 256, 0, stream>>>(labels, i_idx, u_j, counts, rank_,
                                                    start_, s2o, jidx, yall);
    k_normalize<<<B_N, 256, 0, stream>>>(feats, zall, sqn);
    k_slerp<<<NUM_AUG, 256, 0, stream>>>(t_interp, i_idx, jidx, zall, sqn);
    k_ce<<<(N_ALL * 32) / 256, 256, 0, stream>>>(zall, yall, W, b, acc);
    dim3 grid(N_ALL / 64, N_ALL / 32);
    k_gram<<<grid, 256, 0, stream>>>(zall, sqn, yall, acc);
    k_combine<<<1, 256, 0, stream>>>(acc, yall, out);
}